// DecoderAttentionModel_41592463294954
// MI455X (gfx1250) — compile-verified
//
#include <hip/hip_runtime.h>
#include <math.h>

#define BATCH 64
#define ENC_SEQ 2048
#define HID 256
#define WSZ 256
#define DEC_STEPS 128

typedef float v2f __attribute__((ext_vector_type(2)));
typedef float v8f __attribute__((ext_vector_type(8)));

// ---------- device helpers ----------

__device__ __forceinline__ float fast_tanh(float x) {
    // tanh(x) = 1 - 2/(exp(2x)+1); stable for |x| large (-> +/-1)
    float e = __expf(2.0f * x);
    return 1.0f - 2.0f / (e + 1.0f);
}

__device__ __forceinline__ float fast_sigmoid(float x) {
    return 1.0f / (1.0f + __expf(-x));
}

__device__ __forceinline__ float wave_reduce_sum(float v) {
    #pragma unroll
    for (int off = 16; off > 0; off >>= 1)
        v += __shfl_xor(v, off, 32);
    return v;
}

__device__ __forceinline__ float wave_reduce_max(float v) {
    #pragma unroll
    for (int off = 16; off > 0; off >>= 1)
        v = fmaxf(v, __shfl_xor(v, off, 32));
    return v;
}

// ---------- WMMA GEMM: C[m,n] = sum_k A[m,k] * W[n,k] (+ bias0[n] + bias1[n]) ----------
// A: row-major, lda = 256 (K == 256 for all GEMMs here)
// W: row-major (N rows, 256 cols) -> B-matrix B[k][n] = W[n*256 + k]
// One wave computes one 16x16 tile of C using V_WMMA_F32_16X16X4_F32.
__global__ void wmma_gemm256_kernel(const float* __restrict__ A,
                                    const float* __restrict__ W,
                                    const float* __restrict__ bias0,
                                    const float* __restrict__ bias1,
                                    float* __restrict__ C,
                                    int n_tiles /* N/16 */, int N) {
    const int lane = threadIdx.x & 31;
    const int gwave = blockIdx.x * (blockDim.x >> 5) + (threadIdx.x >> 5);
    const int tm = gwave / n_tiles;
    const int tn = gwave - tm * n_tiles;
    const int m0 = tm << 4;
    const int n0 = tn << 4;

    const bool hi = lane >= 16;
    const int l16 = lane & 15;

    // 32-bit A 16x4 fragment layout: lanes 0-15: v0=K+0, v1=K+1 ; lanes 16-31: v0=K+2, v1=K+3
    const float* arow = A + (size_t)(m0 + l16) * 256 + (hi ? 2 : 0);
    // B 4x16 fragment (column n = l16), symmetric layout
    const float* bcol = W + (size_t)(n0 + l16) * 256 + (hi ? 2 : 0);

    v8f acc = {};
    #pragma unroll 8
    for (int k = 0; k < 256; k += 4) {
        v2f a = *reinterpret_cast<const v2f*>(arow + k);   // global_load_b64
        v2f b = *reinterpret_cast<const v2f*>(bcol + k);
        acc = __builtin_amdgcn_wmma_f32_16x16x4_f32(false, a, false, b,
                                                    (short)0, acc, false, false);
    }

    // C/D layout: VGPR r, lanes 0-15 -> M=r, N=lane ; lanes 16-31 -> M=r+8, N=lane-16
    const int row = m0 + (hi ? 8 : 0);
    const int col = n0 + l16;
    float badd = 0.0f;
    if (bias0) badd = bias0[col] + bias1[col];
    #pragma unroll
    for (int r = 0; r < 8; ++r) {
        C[(size_t)(row + r) * N + col] = acc[r] + badd;
    }
}

// ---------- zero init ----------
__global__ void zero_kernel(float* __restrict__ p, int n) {
    int i = blockIdx.x * blockDim.x + threadIdx.x;
    if (i < n) p[i] = 0.0f;
}

// ---------- LSTM cell elementwise: c_new from gates ----------
__global__ void cnew_kernel(const float* __restrict__ gates,
                            const float* __restrict__ enc,
                            float* __restrict__ c) {
    int idx = blockIdx.x * blockDim.x + threadIdx.x;   // 0 .. 64*256-1
    int b = idx >> 8;
    int h = idx & 255;
    const float* g = gates + (size_t)b * (4 * HID);
    float gi = g[h];
    float gf = g[HID + h];
    float gg = g[2 * HID + h];
    float cell0 = enc[((size_t)b * ENC_SEQ + (ENC_SEQ - 1)) * HID + h];
    c[idx] = fast_sigmoid(gf) * cell0 + fast_sigmoid(gi) * fast_tanh(gg);
}

// ---------- attention score: scores[b,s] = sum_w v[w]*tanh(blend1[b,s,w]+blend2[b,w]) ----------
// One wave handles 8 consecutive s rows (same b); v / blend2 cached in registers.
// blend1 reads are coalesced 128B per wave-step; blend1 (128 MiB) stays L2-resident.
__global__ void score_kernel(const float* __restrict__ blend1,
                             const float* __restrict__ blend2,
                             const float* __restrict__ vt,
                             float* __restrict__ scores) {
    const int lane = threadIdx.x & 31;
    const int gwave = blockIdx.x * (blockDim.x >> 5) + (threadIdx.x >> 5);
    const int s0 = gwave * 8;           // global row index (b*2048 + s)
    const int b = s0 >> 11;
    const float* b2row = blend2 + ((size_t)b << 8);

    float vv[8], b2[8];
    #pragma unroll
    for (int k = 0; k < 8; ++k) {
        int w = k * 32 + lane;
        vv[k] = vt[w];
        b2[k] = b2row[w];
    }

    #pragma unroll
    for (int j = 0; j < 8; ++j) {
        const float* b1row = blend1 + (size_t)(s0 + j) * WSZ;
        float p = 0.0f;
        #pragma unroll
        for (int k = 0; k < 8; ++k) {
            float x = b1row[k * 32 + lane] + b2[k];
            p = fmaf(vv[k], fast_tanh(x), p);
        }
        p = wave_reduce_sum(p);
        if (lane == 0) scores[s0 + j] = p;
    }
}

// ---------- log_softmax over s (2048) per batch row, write out[b, t, s] ----------
__global__ void logsoftmax_kernel(const float* __restrict__ scores,
                                  float* __restrict__ out, int t) {
    const int b = blockIdx.x;
    const int tid = threadIdx.x;       // 256 threads
    const int lane = tid & 31;
    const int wid = tid >> 5;
    const float* srow = scores + ((size_t)b << 11);

    float loc[8];
    float mx = -3.402823e38f;
    #pragma unroll
    for (int k = 0; k < 8; ++k) {
        loc[k] = srow[tid + k * 256];
        mx = fmaxf(mx, loc[k]);
    }

    __shared__ float red[8];
    float wm = wave_reduce_max(mx);
    if (lane == 0) red[wid] = wm;
    __syncthreads();
    float bm = red[0];
    #pragma unroll
    for (int k = 1; k < 8; ++k) bm = fmaxf(bm, red[k]);
    __syncthreads();

    float s = 0.0f;
    #pragma unroll
    for (int k = 0; k < 8; ++k) s += __expf(loc[k] - bm);
    float ws = wave_reduce_sum(s);
    if (lane == 0) red[wid] = ws;
    __syncthreads();
    float bs = red[0];
    #pragma unroll
    for (int k = 1; k < 8; ++k) bs += red[k];

    float lse = bm + __logf(bs);
    float* orow = out + ((size_t)b * DEC_STEPS + t) * ENC_SEQ;
    #pragma unroll
    for (int k = 0; k < 8; ++k) orow[tid + k * 256] = loc[k] - lse;
}

// ---------- host launcher ----------
extern "C" void kernel_launch(void* const* d_in, const int* in_sizes, int n_in,
                              void* d_out, int out_size, void* d_ws, size_t ws_size,
                              hipStream_t stream) {
    (void)in_sizes; (void)n_in; (void)out_size; (void)ws_size;
    // setup_inputs order:
    // 0: input (int, unused)  1: encoder_output  2: W_ih (unused)
    // 3: W_hh  4: b_ih  5: b_hh  6: W1  7: W2  8: vt
    const float* enc  = (const float*)d_in[1];
    const float* W_hh = (const float*)d_in[3];
    const float* b_ih = (const float*)d_in[4];
    const float* b_hh = (const float*)d_in[5];
    const float* W1   = (const float*)d_in[6];
    const float* W2   = (const float*)d_in[7];
    const float* vt   = (const float*)d_in[8];
    float* out = (float*)d_out;

    // workspace layout (floats)
    float* blend1 = (float*)d_ws;                                  // 64*2048*256
    float* c      = blend1 + (size_t)BATCH * ENC_SEQ * WSZ;        // 64*256
    float* gates  = c + BATCH * HID;                               // 64*1024
    float* blend2 = gates + BATCH * 4 * HID;                       // 64*256
    float* scores = blend2 + BATCH * WSZ;                          // 64*2048

    const int WAVES_PER_BLOCK = 8;   // 256 threads, wave32

    // blend1 = enc @ W1^T : M = 64*2048, N = 256, K = 256
    {
        int tiles = (BATCH * ENC_SEQ / 16) * (WSZ / 16);           // 131072
        wmma_gemm256_kernel<<<tiles / WAVES_PER_BLOCK, 256, 0, stream>>>(
            enc, W1, nullptr, nullptr, blend1, WSZ / 16, WSZ);
    }
    // carry c = 0
    zero_kernel<<<(BATCH * HID) / 256, 256, 0, stream>>>(c, BATCH * HID);

    for (int t = 0; t < DEC_STEPS; ++t) {
        // gates = (b_ih + b_hh) + c_prev @ W_hh^T : M=64, N=1024, K=256
        {
            int tiles = (BATCH / 16) * (4 * HID / 16);             // 256
            wmma_gemm256_kernel<<<tiles / WAVES_PER_BLOCK, 256, 0, stream>>>(
                c, W_hh, b_ih, b_hh, gates, (4 * HID) / 16, 4 * HID);
        }
        // c_new elementwise
        cnew_kernel<<<(BATCH * HID) / 256, 256, 0, stream>>>(gates, enc, c);
        // blend2 = c_new @ W2^T : M=64, N=256, K=256
        {
            int tiles = (BATCH / 16) * (WSZ / 16);                 // 64
            wmma_gemm256_kernel<<<tiles / WAVES_PER_BLOCK, 256, 0, stream>>>(
                c, W2, nullptr, nullptr, blend2, WSZ / 16, WSZ);
        }
        // scores (dominant pass; blend1 streamed from L2)
        {
            int waves = (BATCH * ENC_SEQ) / 8;                     // 16384 waves
            score_kernel<<<waves / WAVES_PER_BLOCK, 256, 0, stream>>>(
                blend1, blend2, vt, scores);
        }
        // log_softmax + write output slice
        logsoftmax_kernel<<<BATCH, 256, 0, stream>>>(scores, out, t);
    }
}